// GraphConv_67877663146646
// MI455X (gfx1250) — compile-verified
//
#include <hip/hip_runtime.h>
#include <hip/hip_bf16.h>
#include <stdint.h>

#define NV 40000
#define NE 640000
#define DV 128

typedef float v2f __attribute__((ext_vector_type(2)));
typedef float v8f __attribute__((ext_vector_type(8)));

// ---------------------------------------------------------------------------
// 1) zero workspace region (dinv + agg) — harness poisons ws with 0xAA
// ---------------------------------------------------------------------------
__global__ void gcn_zero_kernel(float* __restrict__ ws, int total) {
    int gid = blockIdx.x * blockDim.x + threadIdx.x;
    if (gid < total) ws[gid] = 0.0f;
}

// ---------------------------------------------------------------------------
// 2) in-degree of dst nodes (self-loop added later in rsqrt kernel)
// ---------------------------------------------------------------------------
__global__ void gcn_degree_kernel(const long long* __restrict__ ei,
                                  float* __restrict__ deg) {
    int e = blockIdx.x * blockDim.x + threadIdx.x;   // NE threads exactly
    int dst = (int)ei[NE + e];
    unsafeAtomicAdd(&deg[dst], 1.0f);
}

// ---------------------------------------------------------------------------
// 3) d = rsqrt(deg + 1)
// ---------------------------------------------------------------------------
__global__ void gcn_rsqrt_kernel(float* __restrict__ deg) {
    int i = blockIdx.x * blockDim.x + threadIdx.x;
    if (i < NV) deg[i] = rsqrtf(deg[i] + 1.0f);
}

// ---------------------------------------------------------------------------
// 4) scatter: agg[src] += d[dst] * nodes[dst]
//    One wave (32 lanes) per edge: lane q covers features [4q, 4q+4).
//    Coalesced 512B row read per wave; nodes is L2-resident (20.5MB << 192MB).
// ---------------------------------------------------------------------------
__global__ void gcn_scatter_kernel(const float* __restrict__ nodes,
                                   const long long* __restrict__ ei,
                                   const float* __restrict__ dinv,
                                   float* __restrict__ agg) {
    int gid = blockIdx.x * blockDim.x + threadIdx.x;  // NE*32 threads exactly
    int e = gid >> 5;
    int q = gid & 31;
    int src = (int)ei[e];
    int dst = (int)ei[NE + e];
    float dd = dinv[dst];
    const float4 v = *(const float4*)(nodes + (size_t)dst * DV + 4 * q);
    float* ap = agg + (size_t)src * DV + 4 * q;
    unsafeAtomicAdd(ap + 0, dd * v.x);
    unsafeAtomicAdd(ap + 1, dd * v.y);
    unsafeAtomicAdd(ap + 2, dd * v.z);
    unsafeAtomicAdd(ap + 3, dd * v.w);
}

// ---------------------------------------------------------------------------
// 5) h = d_i * (agg_i + d_i * nodes_i), written in place over agg
// ---------------------------------------------------------------------------
__global__ void gcn_h_kernel(const float* __restrict__ nodes,
                             const float* __restrict__ dinv,
                             float* __restrict__ agg) {
    int gid = blockIdx.x * blockDim.x + threadIdx.x;  // NV*DV/4 threads exactly
    int i = gid >> 5;          // node
    int q = gid & 31;          // float4 slot
    float di = dinv[i];
    size_t idx = (size_t)i * DV + 4 * q;
    float4 a = *(const float4*)(agg + idx);
    float4 n = *(const float4*)(nodes + idx);
    float4 h;
    h.x = di * (a.x + di * n.x);
    h.y = di * (a.y + di * n.y);
    h.z = di * (a.z + di * n.z);
    h.w = di * (a.w + di * n.w);
    *(float4*)(agg + idx) = h;
}

// ---------------------------------------------------------------------------
// 6) out = relu(h @ W^T + b) via V_WMMA_F32_16X16X4_F32 (exact fp32 math).
//    One wave per 16x16 output tile; K=128 -> 32 WMMA per tile.
//    8 waves/block share one 16-row h stripe (all 8 N-tiles). 2500 blocks.
//    Exact grid, no divergence: EXEC is all 1s as WMMA requires.
//
//    A 16x4 f32 layout: lane L holds A[M=L%16][K = (L/16)*2 + {0,1}]
//    B 4x16  f32 layout: lane L holds B[K = (L/16)*2 + {0,1}][N=L%16]
//    C/D 16x16 f32: VGPR r, lane L -> M = r + (L/16)*8, N = L%16
// ---------------------------------------------------------------------------
__global__ void __launch_bounds__(256)
gcn_gemm_wmma_kernel(const float* __restrict__ h,
                     const float* __restrict__ W,
                     const float* __restrict__ bias,
                     float* __restrict__ out) {
    const int lane = threadIdx.x & 31;
    const int wave = threadIdx.x >> 5;           // 0..7
    const int tile = blockIdx.x * 8 + wave;      // 0..19999
    const int tm = tile >> 3;                    // 0..2499  (M tile)
    const int tn = tile & 7;                     // 0..7     (N tile)
    const int m0 = tm * 16;
    const int n0 = tn * 16;

    const int row  = lane & 15;                  // A row / B,D col
    const int koff = (lane >> 4) * 2;            // K sub-offset {0,2}

    const float* hrow = h + (size_t)(m0 + row) * DV;   // A: h[m0+row][*]
    const float* wrow = W + (size_t)(n0 + row) * DV;   // B[k][n] = W[n][k]

    v8f c = {};
#pragma unroll
    for (int k0 = 0; k0 < DV; k0 += 4) {
        v2f a = *(const v2f*)(hrow + k0 + koff);
        v2f b = *(const v2f*)(wrow + k0 + koff);
        // (neg_a, A, neg_b, B, c_mod, C, reuse_a, reuse_b)
        c = __builtin_amdgcn_wmma_f32_16x16x4_f32(
                false, a, false, b, (short)0, c, false, false);
    }

    const float bv = bias[n0 + row];
    const int mbase = m0 + (lane >> 4) * 8;
#pragma unroll
    for (int r = 0; r < 8; ++r) {
        float v = c[r] + bv;
        v = v > 0.0f ? v : 0.0f;
        out[(size_t)(mbase + r) * DV + n0 + row] = v;
    }
}

// ---------------------------------------------------------------------------
extern "C" void kernel_launch(void* const* d_in, const int* in_sizes, int n_in,
                              void* d_out, int out_size, void* d_ws, size_t ws_size,
                              hipStream_t stream) {
    const float*      nodes = (const float*)d_in[0];
    const long long*  ei    = (const long long*)d_in[1];  // int64 (2, NE)
    const float*      W     = (const float*)d_in[2];
    const float*      b     = (const float*)d_in[3];
    float*            out   = (float*)d_out;

    float* dinv = (float*)d_ws;          // NV floats
    float* agg  = dinv + NV;             // NV*DV floats (agg, then h in place)

    const int zero_total = NV + NV * DV;
    gcn_zero_kernel<<<(zero_total + 255) / 256, 256, 0, stream>>>(dinv, zero_total);

    gcn_degree_kernel<<<NE / 256, 256, 0, stream>>>(ei, dinv);

    gcn_rsqrt_kernel<<<(NV + 255) / 256, 256, 0, stream>>>(dinv);

    gcn_scatter_kernel<<<(NE * 32) / 256, 256, 0, stream>>>(nodes, ei, dinv, agg);

    gcn_h_kernel<<<(NV * DV / 4) / 256, 256, 0, stream>>>(nodes, dinv, agg);

    gcn_gemm_wmma_kernel<<<2500, 256, 0, stream>>>(agg, W, b, out);
}